// SymmetricContraction_72318659330811
// MI455X (gfx1250) — compile-verified
//
#include <hip/hip_runtime.h>

// ---------------------------------------------------------------------------
// Symmetric contraction (MACE-style), B=4096 nodes, C=128 channels, I=9, E=10.
// Strategy: per-element fused UW tensors + WMMA_F32_16x16x4 for the Σ_i3 step.
// M-rows of UW3/UW2 are PERMUTED so tile slots j and j+8 share the same i2:
// the stage-2 scatter then needs no selects at all (constant xv index, LDS
// immediate offsets), streaming ds_add_f32 beside the XDL WMMA pipe.
// ---------------------------------------------------------------------------

#define B_N   4096
#define C_N   128
#define E_N   10
#define NT    256      // node tiles (16 nodes each) per element bucket (worst case)
#define MT    21       // m tiles of 16 rows covering 324 rows (padded to 336)
#define MROWS 336
#define KPAD  12       // i3 padded 9 -> 12 (3 WMMA K=4 steps)
#define NPAIR 162      // 9 i2-classes * 18 pairs

typedef __attribute__((ext_vector_type(2))) float v2f;
typedef __attribute__((ext_vector_type(8))) float v8f;

// storage slot -> original m row (or -1 for padding).
// slot = t*16 + h*8 + j ; pair q = t*8 + j ; i2-class = q/18 ;
// within-class row r = 2*(q%18) + h ; note r == stage-2 output row r2.
__device__ __forceinline__ constexpr int sc_slot_m(int slot) {
    int t = slot / 16, sj = slot % 16;
    int h = sj / 8, j = sj % 8;
    int q = t * 8 + j;
    if (q >= NPAIR) return -1;
    int i2 = q / 18;
    int r  = 2 * (q % 18) + h;
    return (r < 9) ? (r * 9 + i2)             // scalar tag: m = i1*9 + i2, i1 = r
                   : (81 + (r - 9) * 9 + i2); // vector tag: m = 81 + (d*9+i1)*9 + i2
}

// ---------------- kernel 0: zero per-element counters ----------------------
__global__ void sc_zero_counts(int* counts) {
    if (threadIdx.x < E_N) counts[threadIdx.x] = 0;
}

// ---------------- kernel 1: bucket nodes by element ------------------------
__global__ void sc_bucket(const float* __restrict__ y, int* counts, int* order) {
    int b = blockIdx.x * blockDim.x + threadIdx.x;
    if (b >= B_N) return;
    int e = 0;
#pragma unroll
    for (int j = 0; j < E_N; ++j)
        if (y[b * E_N + j] > 0.5f) e = j;
    int pos = atomicAdd(&counts[e], 1);
    order[e * B_N + pos] = b;
}

// ---------------- kernel 2: fuse W into U (per element, per channel) -------
// UW3[e][c][slot<336][i3<12], UW2[e][c][slot<336], UW1[e][c][r<36];
// rows stored in permuted slot order (sc_slot_m); padding zeroed.
__global__ void sc_fuse(const float* __restrict__ U2s, const float* __restrict__ U3s,
                        const float* __restrict__ W2s, const float* __restrict__ W3s,
                        const float* __restrict__ U2v, const float* __restrict__ U3v,
                        const float* __restrict__ W2v, const float* __restrict__ W3v,
                        const float* __restrict__ U1s, const float* __restrict__ W1s,
                        const float* __restrict__ U1v, const float* __restrict__ W1v,
                        float* __restrict__ UW3, float* __restrict__ UW2,
                        float* __restrict__ UW1) {
    const size_t T1 = (size_t)E_N * C_N * MROWS * KPAD;
    const size_t T2 = (size_t)E_N * C_N * MROWS;
    const size_t T3 = (size_t)E_N * C_N * 36;
    size_t idx = (size_t)blockIdx.x * blockDim.x + threadIdx.x;
    if (idx < T1) {
        int i3 = (int)(idx % KPAD);
        size_t tq = idx / KPAD;
        int slot = (int)(tq % MROWS);
        int ec   = (int)(tq / MROWS);
        int e = ec / C_N, c = ec % C_N;
        int m = sc_slot_m(slot);
        float s = 0.f;
        if (m >= 0 && i3 < 9) {
            if (m < 81) {
                for (int k = 0; k < 23; ++k)
                    s += U3s[(m * 9 + i3) * 23 + k] * W3s[(e * 23 + k) * C_N + c];
            } else {
                int mm = m - 81;
                for (int k = 0; k < 15; ++k)
                    s += U3v[(mm * 9 + i3) * 15 + k] * W3v[(e * 15 + k) * C_N + c];
            }
        }
        UW3[idx] = s;
    } else if (idx < T1 + T2) {
        size_t j = idx - T1;
        int slot = (int)(j % MROWS);
        int ec   = (int)(j / MROWS);
        int e = ec / C_N, c = ec % C_N;
        int m = sc_slot_m(slot);
        float s = 0.f;
        if (m >= 0) {
            if (m < 81) {
                for (int k = 0; k < 3; ++k)
                    s += U2s[m * 3 + k] * W2s[(e * 3 + k) * C_N + c];
            } else {
                int mm = m - 81;
                for (int k = 0; k < 4; ++k)
                    s += U2v[mm * 4 + k] * W2v[(e * 4 + k) * C_N + c];
            }
        }
        UW2[j] = s;
    } else if (idx < T1 + T2 + T3) {
        size_t j = idx - T1 - T2;
        int r  = (int)(j % 36);
        int ec = (int)(j / 36);
        int e = ec / C_N, c = ec % C_N;
        float s;
        if (r < 9) s = U1s[r] * W1s[e * C_N + c];           // U1_s[9,1]
        else       s = U1v[r - 9] * W1v[e * C_N + c];       // U1_v[3,9,1] flat d*9+i1
        UW1[j] = s;
    }
}

// ---------------- kernel 3: main contraction -------------------------------
// grid = (E_N*NT, C_N/8), block = 256 (8 waves). Wave w handles channel
// c = blockIdx.y*8 + w for a tile of 16 nodes of element e.
__global__ __launch_bounds__(256) void sc_main(const float* __restrict__ x,
                                               const int* __restrict__ counts,
                                               const int* __restrict__ order,
                                               const float* __restrict__ UW3,
                                               const float* __restrict__ UW2,
                                               const float* __restrict__ UW1,
                                               float* __restrict__ out) {
    int e = blockIdx.x / NT;
    int t = blockIdx.x % NT;
    int cnt = counts[e];
    if (t * 16 >= cnt) return;              // uniform across block

    int tid  = threadIdx.x;
    int w    = tid >> 5;
    int lane = tid & 31;
    int c    = blockIdx.y * 8 + w;
    int n    = lane & 15;                   // node column within tile
    bool hi  = lane >= 16;
    int p    = t * 16 + n;
    int node = (p < cnt) ? order[e * B_N + p] : -1;

    __shared__ float s_out2[8][36 * 16];    // wave-private stage-2 accumulators
    __shared__ float s_xt[8][16 * KPAD];    // wave-private x tiles (K-padded)
    float* out2 = s_out2[w];
    float* xt   = s_xt[w];

    const float* A_base   = UW3 + (size_t)(e * C_N + c) * MROWS * KPAD;
    const float* uw2_base = UW2 + (size_t)(e * C_N + c) * MROWS;
    const float* uw1_base = UW1 + (size_t)(e * C_N + c) * 36;

    // warm L2/WGP$ for this wave's UW3 panel (global_prefetch_b8)
    __builtin_prefetch(A_base + (size_t)lane * (MROWS * KPAD / 32), 0, 1);
    __builtin_prefetch(uw1_base, 0, 1);

    // stage x tile into LDS (lanes 0..15: one node each), zero-pad i3 9..11
    if (!hi) {
        const float* xp = x + ((size_t)(node < 0 ? 0 : node) * C_N + c) * 9;
#pragma unroll
        for (int i = 0; i < KPAD; ++i)
            xt[n * KPAD + i] = (i < 9 && node >= 0) ? xp[i] : 0.f;
    }
    for (int i = lane; i < 36 * 16; i += 32) out2[i] = 0.f;
    __syncthreads();

    // per-lane copy of this node-column's x (no ds_loads inside the main loop)
    float xv[9];
#pragma unroll
    for (int i = 0; i < 9; ++i) xv[i] = xt[n * KPAD + i];

    // B operand (x tile) for the 3 K-steps, ISA layout: lo half K+0/K+1, hi K+2/K+3
    const int koff = hi ? 2 : 0;
    v2f bop[3];
#pragma unroll
    for (int kk = 0; kk < 3; ++kk)
        bop[kk] = *(const v2f*)(xt + n * KPAD + kk * 4 + koff);

    // hi-dependent offsets hoisted so the unrolled bodies use pure immediates
    const float* Ap   = A_base + (size_t)n * KPAD + koff;    // + mt*16*KPAD + kk*4
    const float* u2p  = uw2_base + (hi ? 8 : 0);             // + mt*16 + j
    float* out2n = out2 + n + (hi ? 16 : 0);                 // row r2 = 2*(q%18)+h

#pragma unroll
    for (int mt = 0; mt < MT; ++mt) {
        v8f acc = {};
#pragma unroll
        for (int kk = 0; kk < 3; ++kk) {
            // A operand: M(slot) = mt*16 + lane%16, K pair selected by half-wave
            v2f a = *(const v2f*)(Ap + mt * 16 * KPAD + kk * 4);
            acc = __builtin_amdgcn_wmma_f32_16x16x4_f32(
                false, a, false, bop[kk], (short)0, acc, false, false);
        }
        // stage 2: c_t = out3 + UW2; contract over i2 into out2[r2][n].
        // Slot permutation guarantees lo/hi halves share i2 (constant register
        // read) and r2 differs only by +h (folded into out2n). The ds_add_f32
        // offset is a compile-time immediate: zero per-element address math.
#pragma unroll
        for (int j = 0; j < 8; ++j) {
            int q = mt * 8 + j;
            if (q < NPAIR) {
                int i2  = q / 18;             // compile-time
                int off = 2 * (q % 18) * 16;  // compile-time, <= 544 elements
                float ct = acc[j] + u2p[mt * 16 + j];
                atomicAdd(out2n + off, ct * xv[i2]);   // ds_add_f32, imm offset
            }
        }
    }
    __syncthreads();   // all ds atomics visible (uniform control flow)

    // stage 3: c_t1 = out2 + UW1; contract over i1; store final outputs.
#pragma unroll
    for (int half = 0; half < 2; ++half) {
        int it = lane + half * 32;          // 0..63 = 16 nodes x 4 outputs
        int nn = it & 15;
        int o  = it >> 4;                   // 0 = scalar, 1..3 = vector d=o-1
        int pp = t * 16 + nn;
        if (pp < cnt) {
            float s = 0.f;
#pragma unroll
            for (int i1 = 0; i1 < 9; ++i1) {
                int r = (o == 0) ? i1 : (9 + (o - 1) * 9 + i1);
                s += (out2[r * 16 + nn] + uw1_base[r]) * xt[nn * KPAD + i1];
            }
            int nd  = order[e * B_N + pp];
            int col = (o == 0) ? c : (128 + c * 3 + (o - 1));
            out[(size_t)nd * 512 + col] = s;
        }
    }
}

// ---------------------------------------------------------------------------
extern "C" void kernel_launch(void* const* d_in, const int* in_sizes, int n_in,
                              void* d_out, int out_size, void* d_ws, size_t ws_size,
                              hipStream_t stream) {
    // input order (setup_inputs): x, y, U1_s,U2_s,U3_s, W1_s,W2_s,W3_s,
    //                             U1_v,U2_v,U3_v, W1_v,W2_v,W3_v
    const float* x   = (const float*)d_in[0];
    const float* y   = (const float*)d_in[1];
    const float* U1s = (const float*)d_in[2];
    const float* U2s = (const float*)d_in[3];
    const float* U3s = (const float*)d_in[4];
    const float* W1s = (const float*)d_in[5];
    const float* W2s = (const float*)d_in[6];
    const float* W3s = (const float*)d_in[7];
    const float* U1v = (const float*)d_in[8];
    const float* U2v = (const float*)d_in[9];
    const float* U3v = (const float*)d_in[10];
    const float* W1v = (const float*)d_in[11];
    const float* W2v = (const float*)d_in[12];
    const float* W3v = (const float*)d_in[13];
    float* out = (float*)d_out;

    // workspace layout (~22.7 MB): counts | order | UW3 | UW2 | UW1
    int*   counts = (int*)d_ws;
    int*   order  = counts + 64;                           // 256B offset
    float* UW3    = (float*)(order + (size_t)E_N * B_N);   // 16B aligned
    float* UW2    = UW3 + (size_t)E_N * C_N * MROWS * KPAD;
    float* UW1    = UW2 + (size_t)E_N * C_N * MROWS;

    sc_zero_counts<<<1, 64, 0, stream>>>(counts);
    sc_bucket<<<(B_N + 255) / 256, 256, 0, stream>>>(y, counts, order);

    size_t total = (size_t)E_N * C_N * (MROWS * KPAD + MROWS + 36);
    sc_fuse<<<(int)((total + 255) / 256), 256, 0, stream>>>(
        U2s, U3s, W2s, W3s, U2v, U3v, W2v, W3v, U1s, W1s, U1v, W1v,
        UW3, UW2, UW1);

    dim3 grid(E_N * NT, C_N / 8);
    sc_main<<<grid, 256, 0, stream>>>(x, counts, order, UW3, UW2, UW1, out);
}